// SimpleCausticDetector_51960514347331
// MI455X (gfx1250) — compile-verified
//
#include <hip/hip_runtime.h>
#include <hip/hip_bf16.h>
#include <math.h>

#define B_  64
#define T_  4096
#define D_  512
#define DF_ 128
#define TC  16
#define CH  (T_/TC)   // 256; chunks 0..7 are "early" (t<2048), 8..15 "late"

typedef __attribute__((ext_vector_type(2))) float v2f;
typedef __attribute__((ext_vector_type(8))) float v8f;

// ---------------- Pass 1: per-(b,chunk) partial sums / sumsq / masked max ----
__global__ void __launch_bounds__(128)
pass1_kernel(const float* __restrict__ x, const unsigned char* __restrict__ mask,
             float* __restrict__ pS, float* __restrict__ pQ, float* __restrict__ pM) {
    const int tc = blockIdx.x, b = blockIdx.y, tid = threadIdx.x;   // 128 threads, 4 ch each
    __shared__ float vmask[CH];
    const unsigned char* mb = mask + (size_t)b * T_ + (size_t)tc * CH;
    vmask[tid]       = mb[tid]       ? 0.f : 1.f;
    vmask[tid + 128] = mb[tid + 128] ? 0.f : 1.f;
    __syncthreads();

    const float4* xb = (const float4*)(x + ((size_t)b * T_ + (size_t)tc * CH) * D_);
    float s0=0.f,s1=0.f,s2=0.f,s3=0.f, q0=0.f,q1=0.f,q2=0.f,q3=0.f, m=-65000.f;
    #pragma unroll 4
    for (int t = 0; t < CH; ++t) {
        const float v = vmask[t];
        const float4 xv = xb[t * (D_/4) + tid];
        s0 += xv.x*v; q0 += xv.x*xv.x*v;
        s1 += xv.y*v; q1 += xv.y*xv.y*v;
        s2 += xv.z*v; q2 += xv.z*xv.z*v;
        s3 += xv.w*v; q3 += xv.w*xv.w*v;
        const float mm = fmaxf(fmaxf(xv.x, xv.y), fmaxf(xv.z, xv.w));
        m = fmaxf(m, (v != 0.f) ? mm : -65000.f);
    }
    const size_t base = ((size_t)b * TC + tc) * D_ + tid * 4;
    pS[base+0]=s0; pS[base+1]=s1; pS[base+2]=s2; pS[base+3]=s3;
    pQ[base+0]=q0; pQ[base+1]=q1; pQ[base+2]=q2; pQ[base+3]=q3;

    __shared__ float red[128];
    red[tid] = m; __syncthreads();
    for (int s = 64; s > 0; s >>= 1) {
        if (tid < s) red[tid] = fmaxf(red[tid], red[tid + s]);
        __syncthreads();
    }
    if (tid == 0) pM[b * TC + tc] = red[0];
}

// ---------------- Valid-count per batch ----------------
__global__ void __launch_bounds__(256)
denom_kernel(const unsigned char* __restrict__ mask, float* __restrict__ cnt) {
    const int b = blockIdx.x, tid = threadIdx.x;
    float c = 0.f;
    for (int t = tid; t < T_; t += 256) c += mask[(size_t)b * T_ + t] ? 0.f : 1.f;
    __shared__ float red[256];
    red[tid] = c; __syncthreads();
    for (int s = 128; s > 0; s >>= 1) {
        if (tid < s) red[tid] += red[tid + s];
        __syncthreads();
    }
    if (tid == 0) cnt[b] = red[0];
}

// ---------------- Stats: mean/var/early/late maxes -> features 0,1,3 + thr ---
__global__ void __launch_bounds__(512)
stats_kernel(const float* __restrict__ pS, const float* __restrict__ pQ,
             const float* __restrict__ pM, const float* __restrict__ cnt,
             float* __restrict__ feat, float* __restrict__ thr) {
    const int b = blockIdx.x, d = threadIdx.x;   // 512 threads, one per channel
    float e = 0.f, l = 0.f, q = 0.f;
    for (int tc = 0; tc < TC; ++tc) {
        const float sv = pS[((size_t)b * TC + tc) * D_ + d];
        if (tc < TC/2) e += sv; else l += sv;
        q += pQ[((size_t)b * TC + tc) * D_ + d];
    }
    const float s     = e + l;
    const float c     = cnt[b];
    const float denom = c + 1e-8f;
    const float mean  = s / denom;
    const float var   = (q - 2.f * mean * s + mean * mean * c) / denom;

    __shared__ float red[512];
    auto bmax = [&](float v) -> float {
        red[d] = v; __syncthreads();
        for (int st = 256; st > 0; st >>= 1) {
            if (d < st) red[d] = fmaxf(red[d], red[d + st]);
            __syncthreads();
        }
        const float r = red[0]; __syncthreads();
        return r;
    };
    const float vmax = bmax(var);
    const float mmax = bmax(mean);
    const float emax = bmax(e);
    const float lmax = bmax(l);
    if (d == 0) {
        float ms = -65000.f;
        for (int tc = 0; tc < TC; ++tc) ms = fmaxf(ms, pM[b * TC + tc]);
        feat[b*4 + 0] = ms;
        feat[b*4 + 1] = vmax;
        feat[b*4 + 3] = fabsf(emax - lmax);
        thr[b] = 0.7f * mmax;
    }
}

// ---------------- Pass 2: peak counts above threshold ----------------
__global__ void __launch_bounds__(128)
pass2_kernel(const float* __restrict__ x, const unsigned char* __restrict__ mask,
             const float* __restrict__ thr, float* __restrict__ pC) {
    const int tc = blockIdx.x, b = blockIdx.y, tid = threadIdx.x;
    const float th = thr[b];
    __shared__ float vmask[CH];
    const unsigned char* mb = mask + (size_t)b * T_ + (size_t)tc * CH;
    vmask[tid]       = mb[tid]       ? 0.f : 1.f;
    vmask[tid + 128] = mb[tid + 128] ? 0.f : 1.f;
    __syncthreads();

    const float4* xb = (const float4*)(x + ((size_t)b * T_ + (size_t)tc * CH) * D_);
    float c0=0.f,c1=0.f,c2=0.f,c3=0.f;
    #pragma unroll 4
    for (int t = 0; t < CH; ++t) {
        const float v = vmask[t];
        const float4 xv = xb[t * (D_/4) + tid];
        if (xv.x > th) c0 += v;
        if (xv.y > th) c1 += v;
        if (xv.z > th) c2 += v;
        if (xv.w > th) c3 += v;
    }
    const size_t base = ((size_t)b * TC + tc) * D_ + tid * 4;
    pC[base+0]=c0; pC[base+1]=c1; pC[base+2]=c2; pC[base+3]=c3;
}

// ---------------- Peak reduce -> feature 2 ----------------
__global__ void __launch_bounds__(512)
peak_kernel(const float* __restrict__ pC, float* __restrict__ feat) {
    const int b = blockIdx.x, d = threadIdx.x;
    float c = 0.f;
    for (int tc = 0; tc < TC; ++tc) c += pC[((size_t)b * TC + tc) * D_ + d];
    __shared__ float red[512];
    red[d] = c; __syncthreads();
    for (int st = 256; st > 0; st >>= 1) {
        if (d < st) red[d] = fmaxf(red[d], red[d + st]);
        __syncthreads();
    }
    if (d == 0) feat[b*4 + 2] = red[0];
}

// ---------------- MLP: WMMA f32 16x16x4 GEMMs + LN + exact GELU --------------
// One block of 256 threads = 8 waves (wave32). Wave w owns column tile ni=w.
__global__ void __launch_bounds__(256)
mlp_kernel(const float* __restrict__ feat,
           const float* __restrict__ W1, const float* __restrict__ b1,
           const float* __restrict__ gamma, const float* __restrict__ beta,
           const float* __restrict__ W2, const float* __restrict__ b2,
           float* __restrict__ out) {
    __shared__ float hbuf[B_ * DF_];                   // 64x128 = 32 KB
    const int tid = threadIdx.x;
    const int w = tid >> 5, lane = tid & 31;
    const int lm = lane & 15, hi = lane >> 4;
    const int col = w * 16 + lm;

    // ---- GEMM1: h = feat[64,4] @ W1[4,128] + b1  (single K=4 WMMA per tile)
    {
        v2f bf;                                         // B 4x16: lanes0-15 K=0,1; lanes16-31 K=2,3
        bf.x = W1[(2*hi + 0) * DF_ + col];
        bf.y = W1[(2*hi + 1) * DF_ + col];
        const float bias = b1[col];
        #pragma unroll
        for (int mi = 0; mi < 4; ++mi) {
            v2f af;                                     // A 16x4: lane lm = row, v0/v1 = K=2*hi,2*hi+1
            const int row = mi * 16 + lm;
            af.x = feat[row * 4 + 2*hi + 0];
            af.y = feat[row * 4 + 2*hi + 1];
            v8f cf;
            #pragma unroll
            for (int r = 0; r < 8; ++r) cf[r] = bias;
            v8f df = __builtin_amdgcn_wmma_f32_16x16x4_f32(
                false, af, false, bf, (short)0, cf, false, false);
            #pragma unroll
            for (int r = 0; r < 8; ++r)
                hbuf[(mi * 16 + r + 8*hi) * DF_ + col] = df[r];
        }
    }
    __syncthreads();

    // ---- LayerNorm (eps 1e-5) + exact GELU, one thread per row
    if (tid < B_) {
        float* hr = hbuf + tid * DF_;
        float mu = 0.f;
        for (int j = 0; j < DF_; ++j) mu += hr[j];
        mu *= (1.f / DF_);
        float var = 0.f;
        for (int j = 0; j < DF_; ++j) { const float dd = hr[j] - mu; var += dd * dd; }
        var *= (1.f / DF_);
        const float inv = 1.f / sqrtf(var + 1e-5f);
        for (int j = 0; j < DF_; ++j) {
            float hv = (hr[j] - mu) * inv * gamma[j] + beta[j];
            hv = 0.5f * hv * (1.f + erff(hv * 0.70710678118654752f));   // exact GELU
            hr[j] = hv;
        }
    }
    __syncthreads();

    // ---- GEMM2: out = h[64,128] @ W2[128,128] + b2   (32 K=4 WMMA steps)
    v8f acc[4];
    const float bias2 = b2[col];
    #pragma unroll
    for (int mi = 0; mi < 4; ++mi)
        #pragma unroll
        for (int r = 0; r < 8; ++r) acc[mi][r] = bias2;

    for (int kk = 0; kk < 32; ++kk) {
        const int k0 = kk * 4;
        v2f bf;
        bf.x = W2[(k0 + 2*hi + 0) * DF_ + col];
        bf.y = W2[(k0 + 2*hi + 1) * DF_ + col];
        #pragma unroll
        for (int mi = 0; mi < 4; ++mi) {
            v2f af;
            const int row = mi * 16 + lm;
            af.x = hbuf[row * DF_ + k0 + 2*hi + 0];
            af.y = hbuf[row * DF_ + k0 + 2*hi + 1];
            acc[mi] = __builtin_amdgcn_wmma_f32_16x16x4_f32(
                false, af, false, bf, (short)0, acc[mi], false, false);
        }
    }
    #pragma unroll
    for (int mi = 0; mi < 4; ++mi)
        #pragma unroll
        for (int r = 0; r < 8; ++r)
            out[(mi * 16 + r + 8*hi) * DF_ + col] = acc[mi][r];
}

// ---------------- Launch ----------------
extern "C" void kernel_launch(void* const* d_in, const int* in_sizes, int n_in,
                              void* d_out, int out_size, void* d_ws, size_t ws_size,
                              hipStream_t stream) {
    const float*         x     = (const float*)d_in[0];
    const unsigned char* mask  = (const unsigned char*)d_in[1];   // jax bool -> 1 byte
    const float*         W1    = (const float*)d_in[2];
    const float*         b1    = (const float*)d_in[3];
    const float*         gamma = (const float*)d_in[4];
    const float*         beta  = (const float*)d_in[5];
    const float*         W2    = (const float*)d_in[6];
    const float*         b2    = (const float*)d_in[7];
    float*               out   = (float*)d_out;

    // Workspace layout (floats)
    float* ws   = (float*)d_ws;
    const size_t NP = (size_t)B_ * TC * D_;     // 524288
    float* pS   = ws;                           // partial sums
    float* pQ   = pS + NP;                      // partial sumsq
    float* pC   = pQ + NP;                      // partial peak counts
    float* pM   = pC + NP;                      // partial maxes [B*TC]
    float* cnt  = pM + (size_t)B_ * TC;         // valid counts [B]
    float* thr  = cnt + B_;                     // thresholds [B]
    float* feat = thr + B_;                     // features [B*4]

    pass1_kernel<<<dim3(TC, B_), 128, 0, stream>>>(x, mask, pS, pQ, pM);
    denom_kernel<<<B_, 256, 0, stream>>>(mask, cnt);
    stats_kernel<<<B_, 512, 0, stream>>>(pS, pQ, pM, cnt, feat, thr);
    pass2_kernel<<<dim3(TC, B_), 128, 0, stream>>>(x, mask, thr, pC);
    peak_kernel<<<B_, 512, 0, stream>>>(pC, feat);
    mlp_kernel<<<1, 256, 0, stream>>>(feat, W1, b1, gamma, beta, W2, b2, out);
}